// Model_50053548868188
// MI455X (gfx1250) — compile-verified
//
#include <hip/hip_runtime.h>
#include <hip/hip_bf16.h>

typedef __attribute__((ext_vector_type(2))) float v2f;
typedef __attribute__((ext_vector_type(8))) float v8f;

// --------------------------------------------------------------------------
// fp32 WMMA: D = A(16x4) * B(4x16) + C(16x16), all f32.  8-arg form:
// (neg_a, A, neg_b, B, c_mod, C, reuse_a, reuse_b)
// --------------------------------------------------------------------------
__device__ __forceinline__ v8f wmma_f32_k4(v2f a, v2f b, v8f c) {
    return __builtin_amdgcn_wmma_f32_16x16x4_f32(
        /*neg_a=*/false, a, /*neg_b=*/false, b,
        /*c_mod=*/(short)0, c, /*reuse_a=*/false, /*reuse_b=*/false);
}

// --------------------------------------------------------------------------
// Degree count: cnt[dst[e]] += 1
// --------------------------------------------------------------------------
__global__ void count_deg_kernel(const int* __restrict__ dst,
                                 float* __restrict__ cnt, int E) {
    int e = blockIdx.x * blockDim.x + threadIdx.x;
    if (e < E) atomicAdd(&cnt[dst[e]], 1.0f);
}

// --------------------------------------------------------------------------
// Scatter-add: sum[dst[e]][:] += H[src[e]][:]   (one wave32 per edge)
// --------------------------------------------------------------------------
__global__ void scatter_add_kernel(const float* __restrict__ H,
                                   const int* __restrict__ src,
                                   const int* __restrict__ dst,
                                   float* __restrict__ sum,
                                   int E, int d) {
    int t = blockIdx.x * blockDim.x + threadIdx.x;
    int e = t >> 5;
    int lane = t & 31;
    if (e >= E) return;
    const float* hs = H + (size_t)src[e] * d;
    float* sp = sum + (size_t)dst[e] * d;
    for (int i = lane; i < d; i += 32)
        atomicAdd(&sp[i], hs[i]);
}

// --------------------------------------------------------------------------
// Normalize: sum[i] /= max(cnt[i / d], 1)
// --------------------------------------------------------------------------
__global__ void normalize_kernel(float* __restrict__ sum,
                                 const float* __restrict__ cnt,
                                 int total, int d) {
    int i = blockIdx.x * blockDim.x + threadIdx.x;
    if (i >= total) return;
    sum[i] = sum[i] / fmaxf(cnt[i / d], 1.0f);
}

// --------------------------------------------------------------------------
// Fused SAGE GEMM:  H = act( Xself @ Wself + Xneigh @ Wneigh + bias )
// One wave per 16x16 output tile, K stepped by 4 via V_WMMA_F32_16X16X4_F32.
// blockDim = 256 (8 waves). nTile = blockIdx.y*8 + waveId.
// Requires: rows % 16 == 0, din % 4 == 0, dout % 128 == 0 per grid.y math.
// --------------------------------------------------------------------------
__global__ void sage_gemm_kernel(const float* __restrict__ Xself,
                                 const float* __restrict__ Xneigh,
                                 const float* __restrict__ Wself,
                                 const float* __restrict__ Wneigh,
                                 const float* __restrict__ bias,
                                 float* __restrict__ Hout,
                                 int din, int dout, int applyRelu) {
    const int lane  = threadIdx.x & 31;
    const int wave  = threadIdx.x >> 5;
    const int mTile = blockIdx.x;
    const int nTile = blockIdx.y * 8 + wave;
    const int m     = lane & 15;           // A row in tile; also B/C column
    const int khalf = (lane >> 4) * 2;     // K sub-offset: lanes 0-15 -> 0, 16-31 -> 2
    const int rowBase = mTile * 16;
    const int colBase = nTile * 16;

    const float* xs = Xself  + (size_t)(rowBase + m) * din;
    const float* xn = Xneigh + (size_t)(rowBase + m) * din;

    v8f c = {};
    for (int k0 = 0; k0 < din; k0 += 4) {
        const int k = k0 + khalf;
        // A fragments: contiguous float2 per lane
        v2f aS = *(const v2f*)(xs + k);
        v2f aN = *(const v2f*)(xn + k);
        // B fragments: B[k][col], B[k+1][col]
        v2f bS, bN;
        bS.x = Wself [(size_t)(k    ) * dout + colBase + m];
        bS.y = Wself [(size_t)(k + 1) * dout + colBase + m];
        bN.x = Wneigh[(size_t)(k    ) * dout + colBase + m];
        bN.y = Wneigh[(size_t)(k + 1) * dout + colBase + m];
        c = wmma_f32_k4(aS, bS, c);
        c = wmma_f32_k4(aN, bN, c);
    }

    const float bv = bias[colBase + m];
    const int rowHalf = (lane >> 4) * 8;   // C/D: lanes 16-31 hold M = v+8
    #pragma unroll
    for (int v = 0; v < 8; ++v) {
        float val = c[v] + bv;
        if (applyRelu) val = fmaxf(val, 0.0f);
        Hout[(size_t)(rowBase + v + rowHalf) * dout + colBase + m] = val;
    }
}

// --------------------------------------------------------------------------
// Edge score: out[e] = h[src]·Wp[0:d] + h[dst]·Wp[d:2d] + bp  (wave per edge)
// --------------------------------------------------------------------------
__global__ void edge_score_kernel(const float* __restrict__ H,
                                  const int* __restrict__ src,
                                  const int* __restrict__ dst,
                                  const float* __restrict__ Wp,
                                  const float* __restrict__ bp,
                                  float* __restrict__ out,
                                  int E, int d) {
    int t = blockIdx.x * blockDim.x + threadIdx.x;
    int e = t >> 5;
    int lane = t & 31;
    if (e >= E) return;
    const float* hs = H + (size_t)src[e] * d;
    const float* hd = H + (size_t)dst[e] * d;
    float p = 0.0f;
    for (int i = lane; i < d; i += 32)
        p = fmaf(hs[i], Wp[i], fmaf(hd[i], Wp[d + i], p));
    #pragma unroll
    for (int off = 16; off > 0; off >>= 1)
        p += __shfl_xor(p, off, 32);
    if (lane == 0) out[e] = p + bp[0];
}

// --------------------------------------------------------------------------
// Host launch
// --------------------------------------------------------------------------
extern "C" void kernel_launch(void* const* d_in, const int* in_sizes, int n_in,
                              void* d_out, int out_size, void* d_ws, size_t ws_size,
                              hipStream_t stream) {
    const float* x        = (const float*)d_in[0];
    const int*   msg_src  = (const int*)  d_in[1];
    const int*   msg_dst  = (const int*)  d_in[2];
    const int*   pos_src  = (const int*)  d_in[3];
    const int*   pos_dst  = (const int*)  d_in[4];
    const int*   neg_src  = (const int*)  d_in[5];
    const int*   neg_dst  = (const int*)  d_in[6];
    const float* W1_self  = (const float*)d_in[7];
    const float* W1_neigh = (const float*)d_in[8];
    const float* b1       = (const float*)d_in[9];
    const float* W2_self  = (const float*)d_in[10];
    const float* W2_neigh = (const float*)d_in[11];
    const float* b2       = (const float*)d_in[12];
    const float* Wp       = (const float*)d_in[13];
    const float* bp       = (const float*)d_in[14];
    float* out = (float*)d_out;

    const int DIN  = 128;
    const int DHID = in_sizes[9];   // 256
    const int DOUT = in_sizes[12];  // 128
    const int N    = in_sizes[0] / DIN;  // 50000
    const int E    = in_sizes[1];        // 800000
    const int EP   = in_sizes[3];        // 200000
    const int EN   = in_sizes[5];        // 200000

    // Workspace layout (floats): cnt[N] | nsum[N*DHID] | h1[N*DHID] | h2[N*DOUT]
    float* cnt  = (float*)d_ws;
    float* nsum = cnt  + N;
    float* h1   = nsum + (size_t)N * DHID;
    float* h2   = h1   + (size_t)N * DHID;

    // Deterministic init of accumulators each call
    hipMemsetAsync(cnt,  0, (size_t)N * sizeof(float), stream);
    hipMemsetAsync(nsum, 0, (size_t)N * DHID * sizeof(float), stream);

    // Degree counts (same edge list both layers)
    count_deg_kernel<<<(E + 255) / 256, 256, 0, stream>>>(msg_dst, cnt, E);

    // ---- Layer 1 ----
    scatter_add_kernel<<<((size_t)E * 32 + 255) / 256, 256, 0, stream>>>(
        x, msg_src, msg_dst, nsum, E, DIN);
    normalize_kernel<<<((size_t)N * DIN + 255) / 256, 256, 0, stream>>>(
        nsum, cnt, N * DIN, DIN);
    sage_gemm_kernel<<<dim3(N / 16, DHID / 128), 256, 0, stream>>>(
        x, nsum, W1_self, W1_neigh, b1, h1, DIN, DHID, /*relu=*/1);

    // ---- Layer 2 ----
    hipMemsetAsync(nsum, 0, (size_t)N * DHID * sizeof(float), stream);
    scatter_add_kernel<<<((size_t)E * 32 + 255) / 256, 256, 0, stream>>>(
        h1, msg_src, msg_dst, nsum, E, DHID);
    normalize_kernel<<<((size_t)N * DHID + 255) / 256, 256, 0, stream>>>(
        nsum, cnt, N * DHID, DHID);
    sage_gemm_kernel<<<dim3(N / 16, DOUT / 128), 256, 0, stream>>>(
        h1, nsum, W2_self, W2_neigh, b2, h2, DHID, DOUT, /*relu=*/0);

    // ---- Edge scores ----
    edge_score_kernel<<<((size_t)EP * 32 + 255) / 256, 256, 0, stream>>>(
        h2, pos_src, pos_dst, Wp, bp, out, EP, DOUT);
    edge_score_kernel<<<((size_t)EN * 32 + 255) / 256, 256, 0, stream>>>(
        h2, neg_src, neg_dst, Wp, bp, out + EP, EN, DOUT);

    (void)n_in; (void)out_size; (void)ws_size;
}